// PolynomialLayer_35854386987281
// MI455X (gfx1250) — compile-verified
//
#include <hip/hip_runtime.h>

typedef float v2f __attribute__((ext_vector_type(2)));
typedef float v8f __attribute__((ext_vector_type(8)));

#define N_FEAT      8
#define N_TERMS     494
#define N_TERMS_PAD 512
#define BATCH       131072
#define N_FULL      30     // full 16-term tiles; tile 30 is the 14-term tail

// ---- compile-time exponent matrix E[f][t] = multiplicity of feature f in term t,
// enumerated exactly like Python's combinations_with_replacement, degrees 1..4.
struct ETab { float e[N_FEAT][N_TERMS_PAD]; };

static constexpr ETab makeE() {
    ETab T{};   // zero-init (padding terms stay all-zero exponents)
    int idx = 0;
    for (int a = 0; a < N_FEAT; ++a) {                       // degree 1 (8)
        T.e[a][idx] += 1.0f; ++idx;
    }
    for (int a = 0; a < N_FEAT; ++a)                         // degree 2 (36)
        for (int b = a; b < N_FEAT; ++b) {
            T.e[a][idx] += 1.0f; T.e[b][idx] += 1.0f; ++idx;
        }
    for (int a = 0; a < N_FEAT; ++a)                         // degree 3 (120)
        for (int b = a; b < N_FEAT; ++b)
            for (int c = b; c < N_FEAT; ++c) {
                T.e[a][idx] += 1.0f; T.e[b][idx] += 1.0f; T.e[c][idx] += 1.0f; ++idx;
            }
    for (int a = 0; a < N_FEAT; ++a)                         // degree 4 (330)
        for (int b = a; b < N_FEAT; ++b)
            for (int c = b; c < N_FEAT; ++c)
                for (int d = c; d < N_FEAT; ++d) {
                    T.e[a][idx] += 1.0f; T.e[b][idx] += 1.0f;
                    T.e[c][idx] += 1.0f; T.e[d][idx] += 1.0f; ++idx;
                }
    return T;
}
__constant__ ETab g_E = makeE();

// One 16-row x 16-term tile: 2 K=4 GEMM chunks for log-magnitudes, 2 for
// sign-parity, then per-element exp2 + sign-bit XOR + NT streaming store.
template <bool TAIL>
__device__ __forceinline__ void do_tile(int tt, int m, int hi, int k0, int slab,
                                        v2f A0, v2f A1, v2f S0, v2f S1,
                                        const float* __restrict__ w,
                                        float* __restrict__ out) {
    const int t = tt * 16 + m;
    v2f B0 = { g_E.e[k0 + 0][t], g_E.e[k0 + 1][t] };
    v2f B1 = { g_E.e[k0 + 4][t], g_E.e[k0 + 5][t] };

    v8f c  = {};   // sum of e*log2|x|
    v8f cs = {};   // sum of e*negbit  (exact small integer in f32)
    c  = __builtin_amdgcn_wmma_f32_16x16x4_f32(false, A0, false, B0, (short)0, c,  false, false);
    c  = __builtin_amdgcn_wmma_f32_16x16x4_f32(false, A1, false, B1, (short)0, c,  false, false);
    cs = __builtin_amdgcn_wmma_f32_16x16x4_f32(false, S0, false, B0, (short)0, cs, false, false);
    cs = __builtin_amdgcn_wmma_f32_16x16x4_f32(false, S1, false, B1, (short)0, cs, false, false);

    const bool  ok = TAIL ? (t < N_TERMS) : true;
    const float wt = (!TAIL || ok) ? w[t] : 0.0f;
    // D layout: VGPR r holds (M = r + hi*8, N = lane&15)
    float* p = out + (size_t)(slab * 16 + hi * 8) * N_TERMS + t;
#pragma unroll
    for (int r = 0; r < 8; ++r) {
        float mag  = __builtin_amdgcn_exp2f(c[r]);          // v_exp_f32 = 2^x
        int   sgn  = ((int)cs[r]) << 31;                    // exact cvt; parity -> sign bit
        int   bits = __float_as_int(wt * mag) ^ sgn;
        if (ok)
            __builtin_nontemporal_store(bits, (int*)(p + (size_t)r * N_TERMS));
    }
}

__global__ __launch_bounds__(256)
void poly_wmma_kernel(const float* __restrict__ x,
                      const float* __restrict__ w,
                      float* __restrict__ out) {
    const int lane = threadIdx.x & 31;
    const int wid  = threadIdx.x >> 5;
    const int slab = blockIdx.x * 8 + wid;   // one 16-row slab per wave
    const int m    = lane & 15;              // M index for A, N index for B/D
    const int hi   = lane >> 4;              // lane half selects K pair / M+8
    const int k0   = hi * 2;                 // K base within a K=4 chunk

    // hint the tiny weight table into cache (global_prefetch_b8)
    __builtin_prefetch(w + ((size_t)lane * 64 % N_TERMS), 0, 0);

    // ---- A operands: log2|x| (clamped) and sign bits, per ISA 16x4 f32 A layout:
    // lanes 0-15 hold K = {0,1} in the two VGPRs, lanes 16-31 hold K = {2,3}.
    const float* xr = x + (size_t)(slab * 16 + m) * N_FEAT;
    const float x0 = xr[k0 + 0], x1 = xr[k0 + 1];
    const float x2 = xr[k0 + 4], x3 = xr[k0 + 5];

    auto lg = [](float v) {
        return fmaxf(__builtin_amdgcn_logf(fabsf(v)), -8192.0f); // v_log_f32 = log2
    };
    v2f A0 = { lg(x0), lg(x1) };                                  // K = 0..3 chunk
    v2f A1 = { lg(x2), lg(x3) };                                  // K = 4..7 chunk
    v2f S0 = { x0 < 0.0f ? 1.0f : 0.0f, x1 < 0.0f ? 1.0f : 0.0f };
    v2f S1 = { x2 < 0.0f ? 1.0f : 0.0f, x3 < 0.0f ? 1.0f : 0.0f };

    // 30 full tiles: no bounds checks, EXEC stays all-ones, straight-line stores.
    for (int tt = 0; tt < N_FULL; ++tt)
        do_tile<false>(tt, m, hi, k0, slab, A0, A1, S0, S1, w, out);
    // tail tile (terms 480..493): masked stores only here.
    do_tile<true>(N_FULL, m, hi, k0, slab, A0, A1, S0, S1, w, out);
}

extern "C" void kernel_launch(void* const* d_in, const int* in_sizes, int n_in,
                              void* d_out, int out_size, void* d_ws, size_t ws_size,
                              hipStream_t stream) {
    const float* x = (const float*)d_in[0];        // (131072, 8)
    const float* w = (const float*)d_in[1];        // (494,)
    float* out = (float*)d_out;                    // (131072, 494)
    (void)in_sizes; (void)n_in; (void)out_size; (void)d_ws; (void)ws_size;

    const int blocks = BATCH / (16 * 8);           // 1024 blocks, 8 waves/block
    poly_wmma_kernel<<<blocks, 256, 0, stream>>>(x, w, out);
}